// HintGenKernel_8057358647764
// MI455X (gfx1250) — compile-verified
//
#include <hip/hip_runtime.h>
#include <stdint.h>

// PIR-style hint generation: per hint, XOR-reduce a masked gather of 512
// entries (5 x u64 each) from a 1M x 40B table.
//
// MI455X reasoning:
//  * entries table = 40 MB -> resident in 192 MB L2; the random gather
//    (~671 MB of 40B rows) is L2-bound, not HBM-bound.
//  * index/mask streams (~152 MB) are the only HBM traffic -> stream the
//    index chunks into LDS with gfx1250 async-to-LDS DMA (ASYNCcnt),
//    double-buffered so DMA overlaps the random gathers.
//  * XOR is bitwise -> VALU v_xor_b32 is the right "compute"; WMMA would
//    need bit->byte unpack (8x expansion) and is a de-optimization.
//  * wave32: one wave per hint, butterfly XOR reduce across 32 lanes.

#define WAVES_PER_BLOCK 8
#define CHUNK 128              // subset slots staged per async chunk (1 KB of int64 indices)

__global__ __launch_bounds__(256) void
hint_parity_kernel(const unsigned long long* __restrict__ entries,
                   const unsigned long long* __restrict__ indices,
                   const unsigned char*      __restrict__ valid,
                   unsigned long long*       __restrict__ out,
                   int num_hints, int subset)
{
    // Double-buffered per-wave index staging: 2 x 128 x 8B = 2 KB/wave, 16 KB/block.
    __shared__ unsigned long long idxbuf[WAVES_PER_BLOCK][2][CHUNK];

    const int lane = threadIdx.x & 31;
    const int wave = threadIdx.x >> 5;
    const int hint = blockIdx.x * WAVES_PER_BLOCK + wave;
    if (hint >= num_hints) return;          // wave-uniform exit

    const unsigned long long* src  = indices + (size_t)hint * subset;
    const unsigned char*      msk  = valid   + (size_t)hint * subset;

    // Raw LDS offsets for the async DMA destination (generic LDS pointers on
    // gfx1250 carry the aperture in the high dword; low 32 bits are the LDS
    // byte address).
    const unsigned lds0 = (unsigned)(uintptr_t)&idxbuf[wave][0][0];
    const unsigned lds1 = (unsigned)(uintptr_t)&idxbuf[wave][1][0];

    const int nchunk = subset / CHUNK;

    // Issue one chunk (1 KB = 128 int64 indices) as 2 async-to-LDS b128 ops:
    // 32 lanes x 16B each, instruction offset covers the second 512B half
    // (INST_OFFSET is added to both the LDS and global addresses).
    auto issue_chunk = [&](unsigned ldsbase, const unsigned long long* gsrc) {
        unsigned long long ga = (unsigned long long)(uintptr_t)gsrc
                              + (unsigned long long)(lane * 16);
        unsigned           la = ldsbase + (unsigned)(lane * 16);
        asm volatile(
            "global_load_async_to_lds_b128 %0, %1, off\n\t"
            "global_load_async_to_lds_b128 %0, %1, off offset:512"
            :: "v"(la), "v"(ga) : "memory");
    };

    issue_chunk(lds0, src);                 // prologue: chunk 0 -> buffer 0

    unsigned long long acc0 = 0, acc1 = 0, acc2 = 0, acc3 = 0, acc4 = 0;

    for (int c = 0; c < nchunk; ++c) {
        const int buf = c & 1;

        if (c + 1 < nchunk) {
            issue_chunk(buf ? lds0 : lds1, src + (size_t)(c + 1) * CHUNK);
            // Prefetch next chunk's mask bytes into cache (global_prefetch_b8).
            __builtin_prefetch(msk + (size_t)(c + 1) * CHUNK + lane, 0, 3);
            // 2 ops outstanding for chunk c+1; async ops complete in order, so
            // asynccnt <= 2 means chunk c's data has landed in LDS.
            asm volatile("s_wait_asynccnt 0x2" ::: "memory");
        } else {
            asm volatile("s_wait_asynccnt 0x0" ::: "memory");
        }

        const unsigned char* mchunk = msk + (size_t)c * CHUNK;

        #pragma unroll
        for (int j = 0; j < CHUNK / 32; ++j) {
            const int s = j * 32 + lane;
            // ds_load_b64 from our private staging buffer (value < 2^20).
            const unsigned idx = (unsigned)idxbuf[wave][buf][s];
            // Branchless validity mask: 0x000..0 or 0xFFF..F.
            const unsigned long long m =
                0ULL - (unsigned long long)(mchunk[s] & 1);

            // 40B row, 8B-aligned -> five independent global_load_b64 gathers
            // (L2-resident table); unrolled x4 keeps ~20 loads in flight/lane.
            const unsigned long long* e = entries + (size_t)idx * 5u;
            acc0 ^= e[0] & m;
            acc1 ^= e[1] & m;
            acc2 ^= e[2] & m;
            acc3 ^= e[3] & m;
            acc4 ^= e[4] & m;
        }
    }

    // Butterfly XOR reduction across the wave32 lanes (ds-permute path).
    #pragma unroll
    for (int off = 16; off > 0; off >>= 1) {
        acc0 ^= __shfl_xor(acc0, off, 32);
        acc1 ^= __shfl_xor(acc1, off, 32);
        acc2 ^= __shfl_xor(acc2, off, 32);
        acc3 ^= __shfl_xor(acc3, off, 32);
        acc4 ^= __shfl_xor(acc4, off, 32);
    }

    if (lane == 0) {
        unsigned long long* o = out + (size_t)hint * 5u;
        o[0] = acc0; o[1] = acc1; o[2] = acc2; o[3] = acc3; o[4] = acc4;
    }
}

extern "C" void kernel_launch(void* const* d_in, const int* in_sizes, int n_in,
                              void* d_out, int out_size, void* d_ws, size_t ws_size,
                              hipStream_t stream) {
    (void)n_in; (void)d_ws; (void)ws_size;
    const unsigned long long* entries = (const unsigned long long*)d_in[0]; // int64 [1M,5]
    const unsigned long long* indices = (const unsigned long long*)d_in[1]; // int64 [H,S]
    const unsigned char*      valid   = (const unsigned char*)d_in[2];      // bool  [H,S]
    unsigned long long*       out     = (unsigned long long*)d_out;         // int64 [H,5]

    const int num_hints = out_size / 5;                       // 32768
    const int subset    = num_hints ? in_sizes[1] / num_hints // 512
                                    : 0;
    if (num_hints <= 0 || subset <= 0) return;

    const int blocks = (num_hints + WAVES_PER_BLOCK - 1) / WAVES_PER_BLOCK;
    hint_parity_kernel<<<blocks, 256, 0, stream>>>(entries, indices, valid,
                                                   out, num_hints, subset);
}